// MaxAttention_7928509629210
// MI455X (gfx1250) — compile-verified
//
#include <hip/hip_runtime.h>
#include <hip/hip_bf16.h>
#include <math.h>

#define B_ 128
#define S_ 2048
#define H_ 512
#define L_ 11

typedef __attribute__((ext_vector_type(2))) float v2f;
typedef __attribute__((ext_vector_type(8))) float v8f;

// ---------------------------------------------------------------------------
// Kernel 1 (dominant, HBM-bound: 536 MB streamed): masked max + argmax per
// (b,h) row. Rows are contiguous 2048-float chunks. One wave32 per row,
// float4 loads (512B/wave/iter, fully coalesced), shuffle-based reduction.
// ---------------------------------------------------------------------------
__global__ __launch_bounds__(256) void k_maxpool(const float* __restrict__ x,
                                                 const int* __restrict__ lengths,
                                                 float* __restrict__ pool,
                                                 int* __restrict__ idxout) {
  const int wave = threadIdx.x >> 5;
  const int lane = threadIdx.x & 31;
  const int row  = blockIdx.x * 8 + wave;     // 0 .. B*H-1
  const int b    = row >> 9;                  // row / H_
  const int len  = lengths[b];
  const float* rp = x + (size_t)row * S_;

  float best = -INFINITY;
  int   bidx = 0x7FFFFFFF;
  #pragma unroll
  for (int i = 0; i < 16; ++i) {
    const int s = i * 128 + lane * 4;
    const float4 v = *(const float4*)(rp + s);
    const float vv[4] = {v.x, v.y, v.z, v.w};
    #pragma unroll
    for (int j = 0; j < 4; ++j) {
      const int ss = s + j;
      if (ss < len && vv[j] > best) { best = vv[j]; bidx = ss; }  // strict > keeps first max
    }
  }
  // wave32 cross-lane reduce; tie -> smaller index (jnp.argmax = first)
  #pragma unroll
  for (int off = 16; off > 0; off >>= 1) {
    const float ov = __shfl_xor(best, off, 32);
    const int   oi = __shfl_xor(bidx, off, 32);
    if (ov > best || (ov == best && oi < bidx)) { best = ov; bidx = oi; }
  }
  if (lane == 0) { pool[row] = best; idxout[row] = bidx; }
}

// ---------------------------------------------------------------------------
// Kernel 2: attn = code_pool(128x512) @ fc_w^T(512x11) with
// V_WMMA_F32_16X16X4_F32 (fp32 exact). 1 block, 8 waves; wave w = M-tile w.
// N padded 11 -> 16 with zero columns (masked B operand, no divergence
// around the WMMA so EXEC stays all-ones).
// ---------------------------------------------------------------------------
__global__ __launch_bounds__(256) void k_attn_wmma(const float* __restrict__ pool,
                                                   const float* __restrict__ fcw,
                                                   float* __restrict__ attn) {
  const int wave = threadIdx.x >> 5;   // M tile 0..7
  const int lane = threadIdx.x & 31;
  const int half = lane >> 4;          // 0: K=k..k+1, 1: K=k+2..k+3
  const int lid  = lane & 15;          // A: row m; B: col n

  const float* arow = pool + (size_t)(wave * 16 + lid) * H_;
  const int   nclmp = (lid < L_) ? lid : (L_ - 1);
  const float bmask = (lid < L_) ? 1.0f : 0.0f;
  const float* brow = fcw + (size_t)nclmp * H_;   // B[k,n] = fc_w[n*H + k]

  v8f c = {};
  for (int kk = 0; kk < H_; kk += 4) {
    const int k2 = kk + half * 2;
    v2f a, bb;
    a.x  = arow[k2];
    a.y  = arow[k2 + 1];
    bb.x = brow[k2] * bmask;
    bb.y = brow[k2 + 1] * bmask;
    // 8 args: (neg_a, A, neg_b, B, c_mod, C, reuse_a, reuse_b)
    c = __builtin_amdgcn_wmma_f32_16x16x4_f32(false, a, false, bb,
                                              (short)0, c, false, false);
  }
  // C/D layout: lane n = lid, VGPR v -> M = half*8 + v
  if (lid < L_) {
    const int mbase = wave * 16 + half * 8;
    #pragma unroll
    for (int v = 0; v < 8; ++v)
      attn[(size_t)(mbase + v) * L_ + lid] = c[v];
  }
}

// ---------------------------------------------------------------------------
// Kernel 3: contri_out[:, :, 0:2] = contri ; channel 2 zeroed every launch
// (d_out is not re-poisoned between graph replays; zero-then-scatter).
// ---------------------------------------------------------------------------
__global__ __launch_bounds__(256) void k_copy(const float* __restrict__ contri,
                                              float* __restrict__ out3) {
  const int i = blockIdx.x * 256 + threadIdx.x;     // 0 .. B*S-1
  if (i < B_ * S_) {
    const float2 cv = *(const float2*)(contri + (size_t)i * 2);
    float* o = out3 + (size_t)i * 3;
    o[0] = cv.x;
    o[1] = cv.y;
    o[2] = 0.0f;
  }
}

// ---------------------------------------------------------------------------
// Kernel 4: rowsum[b,l] = sum_h relu(pool[b,h]*fcw[l,h]) + 1e-4, b<10.
// ---------------------------------------------------------------------------
__global__ __launch_bounds__(256) void k_rowsum(const float* __restrict__ pool,
                                                const float* __restrict__ fcw,
                                                float* __restrict__ rowsum) {
  __shared__ float red[256];
  const int bid = blockIdx.x;          // 0..109
  const int b = bid / L_, l = bid % L_;
  float s = 0.0f;
  for (int h = threadIdx.x; h < H_; h += 256)
    s += fmaxf(pool[b * H_ + h] * fcw[l * H_ + h], 0.0f) + 1e-4f;
  red[threadIdx.x] = s;
  __syncthreads();
  for (int off = 128; off > 0; off >>= 1) {
    if (threadIdx.x < off) red[threadIdx.x] += red[threadIdx.x + off];
    __syncthreads();
  }
  if (threadIdx.x == 0) rowsum[bid] = red[0];
}

// ---------------------------------------------------------------------------
// Kernel 5: total[b,h] = sum_l 100*con/rowsum ; atomic scatter into channel 2
// at s = code_indices[b,h]. Runs after k_copy has zeroed channel 2.
// ---------------------------------------------------------------------------
__global__ __launch_bounds__(256) void k_scatter(const float* __restrict__ pool,
                                                 const int* __restrict__ idx,
                                                 const float* __restrict__ fcw,
                                                 const float* __restrict__ rowsum,
                                                 float* __restrict__ out3) {
  const int i = blockIdx.x * 256 + threadIdx.x;   // 0 .. 10*H-1
  const int b = i / H_, h = i % H_;
  const float p = pool[b * H_ + h];
  float total = 0.0f;
  #pragma unroll
  for (int l = 0; l < L_; ++l) {
    const float con = fmaxf(p * fcw[l * H_ + h], 0.0f) + 1e-4f;
    total += con / rowsum[b * L_ + l] * 100.0f;
  }
  const int s = idx[b * H_ + h];
  atomicAdd(out3 + (size_t)b * S_ * 3 + (size_t)s * 3 + 2, total);
}

extern "C" void kernel_launch(void* const* d_in, const int* in_sizes, int n_in,
                              void* d_out, int out_size, void* d_ws, size_t ws_size,
                              hipStream_t stream) {
  // inputs: find_contri(=1, unused), code_output, lengths, contri, fc_w
  const float* code_output = (const float*)d_in[1];
  const int*   lengths     = (const int*)  d_in[2];
  const float* contri      = (const float*)d_in[3];
  const float* fcw         = (const float*)d_in[4];

  float* attn = (float*)d_out;                 // B*L floats
  float* out3 = (float*)d_out + B_ * L_;       // B*S*3 floats

  // workspace: pool(256KB) | indices(256KB) | rowsum(440B)
  float* pool   = (float*)d_ws;
  int*   idxbuf = (int*)((char*)d_ws + (size_t)B_ * H_ * sizeof(float));
  float* rowsum = (float*)((char*)d_ws + 2ull * B_ * H_ * sizeof(float));

  k_maxpool  <<<(B_ * H_) / 8, 256, 0, stream>>>(code_output, lengths, pool, idxbuf);
  k_attn_wmma<<<1, 256, 0, stream>>>(pool, fcw, attn);
  k_copy     <<<(B_ * S_ + 255) / 256, 256, 0, stream>>>(contri, out3);
  k_rowsum   <<<10 * L_, 256, 0, stream>>>(pool, fcw, rowsum);
  k_scatter  <<<(10 * H_) / 256, 256, 0, stream>>>(pool, idxbuf, fcw, rowsum, out3);
}